// SimpleModel_87849261072474
// MI455X (gfx1250) — compile-verified
//
#include <hip/hip_runtime.h>

// Embedding gather: out[row, :] = table[idx[row], :]
//   idx:   819,200 int32 in [0, 1e6)
//   table: 1e6 x 128 float32 (512 MB)
//   out:   819,200 x 128 float32 (419 MB)
//
// Pure HBM-bandwidth problem (~840 MB traffic -> ~36 us @ 23.3 TB/s).
// One wave32 per row: lane l moves float4 -> 512 B/row via b128 ops.
// 8 rows batched per wave for memory-level parallelism; row block index is
// forced scalar (readfirstlane) so index fetch goes to the SMEM/KMcnt path
// and all vector memory ops use scalar-base (saddr) addressing.
// Table loads stay regular-temporal (L2 reuse, ~31% repeat hits in 192 MB L2),
// output stores are non-temporal so the 419 MB write stream doesn't evict
// table lines.

typedef float v4f __attribute__((ext_vector_type(4)));

#define EMB_DIM       128
#define VEC_PER_ROW   (EMB_DIM / 4)   // 32 float4 per row == one wave32
#define ROWS_PER_WAVE 8
#define BLOCK_THREADS 256             // 8 wave32 per block

__global__ __launch_bounds__(BLOCK_THREADS) void emb_gather_kernel(
    const int* __restrict__ idx,
    const float* __restrict__ table,
    float* __restrict__ out,
    int num_rows)
{
    const int lane        = threadIdx.x & 31;           // wave32 on gfx1250
    const int wave_in_blk = threadIdx.x >> 5;           // 0..7

    // Wave-uniform first row, forced into an SGPR so index loads and all
    // memory-op base addresses are scalar.
    int row0 = (blockIdx.x * (BLOCK_THREADS / 32) + wave_in_blk) * ROWS_PER_WAVE;
    row0 = __builtin_amdgcn_readfirstlane(row0);
    if (row0 >= num_rows) return;

    const v4f* __restrict__ t4 = reinterpret_cast<const v4f*>(table);
    v4f* __restrict__       o4 = reinterpret_cast<v4f*>(out);

    if (row0 + ROWS_PER_WAVE <= num_rows) {
        // Uniform, read-only, restrict-qualified -> SMEM (s_load) candidates.
        int ids[ROWS_PER_WAVE];
#pragma unroll
        for (int r = 0; r < ROWS_PER_WAVE; ++r)
            ids[r] = idx[row0 + r];

        // 8 independent 512 B random gathers in flight before any wait.
        v4f v[ROWS_PER_WAVE];
#pragma unroll
        for (int r = 0; r < ROWS_PER_WAVE; ++r)
            v[r] = t4[((size_t)(unsigned)ids[r] * VEC_PER_ROW) + lane];

        // Streaming store path: th:NT keeps L2 free for table rows.
#pragma unroll
        for (int r = 0; r < ROWS_PER_WAVE; ++r)
            __builtin_nontemporal_store(
                v[r], &o4[((size_t)(unsigned)(row0 + r) * VEC_PER_ROW) + lane]);
    } else {
        // Tail (not reached for 819,200 rows: divisible by 8; kept for safety).
        for (int r = 0; r < ROWS_PER_WAVE; ++r) {
            const int row = row0 + r;
            if (row >= num_rows) break;
            const int id = idx[row];
            v4f v = t4[((size_t)(unsigned)id * VEC_PER_ROW) + lane];
            __builtin_nontemporal_store(
                v, &o4[((size_t)(unsigned)row * VEC_PER_ROW) + lane]);
        }
    }
}

extern "C" void kernel_launch(void* const* d_in, const int* in_sizes, int n_in,
                              void* d_out, int out_size, void* d_ws, size_t ws_size,
                              hipStream_t stream)
{
    const int*   x     = (const int*)d_in[0];     // [16384, 50] int32 ids
    const float* table = (const float*)d_in[1];   // [1e6, 128] float32
    float*       out   = (float*)d_out;           // [16384, 50, 128] float32

    const int num_rows = in_sizes[0];             // 819,200 lookups
    const int waves    = (num_rows + ROWS_PER_WAVE - 1) / ROWS_PER_WAVE;
    const int blocks   = (waves * 32 + BLOCK_THREADS - 1) / BLOCK_THREADS;

    emb_gather_kernel<<<blocks, BLOCK_THREADS, 0, stream>>>(x, table, out, num_rows);
}